// Hidden_state_Attention_36996848288270
// MI455X (gfx1250) — compile-verified
//
#include <hip/hip_runtime.h>
#include <hip/hip_bf16.h>
#include <math.h>

// ---- problem constants (from reference) ----
#define LSEQ   2048
#define BDIM   64
#define DDIM   256
#define CCH    512
#define NSPLIT 8          // l-dimension splits for main kernel
#define ROWS_PER_BLOCK (LSEQ / NSPLIT)   // 256

typedef __attribute__((ext_vector_type(2))) float v2f;
typedef __attribute__((ext_vector_type(4))) float v4f;
typedef __attribute__((ext_vector_type(8))) float v8f;

// ---------------------------------------------------------------------------
// Kernel 1: base[b,d] = (ht_query @ W_query^T)[b,d] + b_att_past[d]
// FP32 WMMA 16x16x4. 64 waves total, each owns one 16x16 output tile.
// A (16x4):  lanes 0-15 rows M=0..15, v0=K0/K2, v1=K1/K3 (lane<16 -> K0/K1)
// B (4x16):  lanes hold N=lane%16; v0 = K0 (lanes 0-15) / K2 (16-31), v1 = K1/K3
// C/D (16x16): VGPR r -> M = r + (lane<16 ? 0 : 8), N = lane%16
// ---------------------------------------------------------------------------
__global__ __launch_bounds__(256) void query_gemm(const float* __restrict__ htq,
                                                  const float* __restrict__ Wq,
                                                  const float* __restrict__ b_att,
                                                  float* __restrict__ ws_base) {
  const int lane = threadIdx.x & 31;
  const int wave = threadIdx.x >> 5;
  const int wid  = blockIdx.x * 8 + wave;     // 0..63 tiles
  const int b0 = (wid >> 4) * 16;             // 4 tiles along B
  const int n0 = (wid & 15) * 16;             // 16 tiles along D
  const int m    = lane & 15;
  const int koff = (lane < 16) ? 0 : 2;

  v8f c = {};
  #pragma unroll 4
  for (int e0 = 0; e0 < DDIM; e0 += 4) {
    v2f a, bm;
    a.x  = htq[(b0 + m) * DDIM + e0 + koff];
    a.y  = htq[(b0 + m) * DDIM + e0 + koff + 1];
    bm.x = Wq[(n0 + m) * DDIM + e0 + koff];      // B[k][n] = Wq[n][k] (hq = x @ Wq^T)
    bm.y = Wq[(n0 + m) * DDIM + e0 + koff + 1];
    c = __builtin_amdgcn_wmma_f32_16x16x4_f32(false, a, false, bm, (short)0, c,
                                              false, false);
  }
  const int mbase = (lane < 16) ? 0 : 8;
  const float bias = b_att[n0 + m];
  #pragma unroll
  for (int r = 0; r < 8; ++r)
    ws_base[(size_t)(b0 + mbase + r) * DDIM + n0 + m] = c[r] + bias;
}

// ---------------------------------------------------------------------------
// Kernel 2: fold conv1d weights into fc_att_past:  M[k][d] = sum_c Wap[d,c]*Wc[c,k]
// ---------------------------------------------------------------------------
__global__ __launch_bounds__(256) void prep_M(const float* __restrict__ Wap,
                                              const float* __restrict__ Wc,
                                              float* __restrict__ wsM) {
  const int d = threadIdx.x;
  float m0 = 0.f, m1 = 0.f, m2 = 0.f;
  #pragma unroll 4
  for (int ch = 0; ch < CCH; ++ch) {
    const float w = Wap[d * CCH + ch];
    m0 += w * Wc[ch * 3 + 0];
    m1 += w * Wc[ch * 3 + 1];
    m2 += w * Wc[ch * 3 + 2];
  }
  wsM[0 * DDIM + d] = m0;
  wsM[1 * DDIM + d] = m1;
  wsM[2 * DDIM + d] = m2;
}

// ---------------------------------------------------------------------------
// Kernel 3: streaming score + gated weighted sum.
// grid = (NSPLIT, B); block = 256 (8 waves). Wave processes one (l,b) row at
// a time: 256 d-values = 8 floats/lane (two 128-bit NT loads).
// ---------------------------------------------------------------------------
__device__ __forceinline__ v4f ld4_nt(const float* p) {
  return __builtin_nontemporal_load((const v4f*)p);
}
__device__ __forceinline__ v4f ld4(const float* p) {
  return *(const v4f*)p;
}

__global__ __launch_bounds__(256) void main_kernel(
    const float* __restrict__ ctx_val, const float* __restrict__ ctx_key,
    const float* __restrict__ ctx_mask, const float* __restrict__ AP,
    const float* __restrict__ W_attn, const float* __restrict__ b_attn,
    const float* __restrict__ ws_base, const float* __restrict__ wsM,
    float* __restrict__ ctp, float* __restrict__ Sp,
    float* __restrict__ hard_out) {
  __shared__ float ap_s[ROWS_PER_BLOCK + 2];
  __shared__ float mask_s[ROWS_PER_BLOCK];
  __shared__ float ctLDS[8 * DDIM];
  __shared__ float SLDS[8];

  const int split = blockIdx.x;
  const int b     = blockIdx.y;
  const int lbase = split * ROWS_PER_BLOCK;

  // stage attention_past window (with conv zero-padding) and mask into LDS
  for (int t = threadIdx.x; t < ROWS_PER_BLOCK + 2; t += 256) {
    const int gl = lbase - 1 + t;
    ap_s[t] = (gl >= 0 && gl < LSEQ) ? AP[(size_t)b * LSEQ + gl] : 0.f;
  }
  mask_s[threadIdx.x] = ctx_mask[(size_t)(lbase + threadIdx.x) * BDIM + b];
  __syncthreads();

  const int lane = threadIdx.x & 31;
  const int wave = threadIdx.x >> 5;
  const int d0 = lane * 4;
  const int d1 = 128 + lane * 4;

  const v4f wa0 = ld4(W_attn + d0);
  const v4f wa1 = ld4(W_attn + d1);
  const v4f bs0 = ld4(ws_base + (size_t)b * DDIM + d0);
  const v4f bs1 = ld4(ws_base + (size_t)b * DDIM + d1);
  const v4f m00 = ld4(wsM + 0 * DDIM + d0);
  const v4f m01 = ld4(wsM + 0 * DDIM + d1);
  const v4f m10 = ld4(wsM + 1 * DDIM + d0);
  const v4f m11 = ld4(wsM + 1 * DDIM + d1);
  const v4f m20 = ld4(wsM + 2 * DDIM + d0);
  const v4f m21 = ld4(wsM + 2 * DDIM + d1);
  const float battn = b_attn[0];

  v4f acc0 = {0.f, 0.f, 0.f, 0.f};
  v4f acc1 = {0.f, 0.f, 0.f, 0.f};
  float Sacc = 0.f;

  for (int i = 0; i < ROWS_PER_BLOCK / 8; ++i) {
    const int j = i * 8 + wave;            // row index within block (wave-uniform)
    const int l = lbase + j;
    const float am1 = ap_s[j];             // AP[l-1]
    const float a0  = ap_s[j + 1];         // AP[l]
    const float ap1 = ap_s[j + 2];         // AP[l+1]

    const float* keyrow = ctx_key + ((size_t)l * BDIM + b) * DDIM;
    if (i + 1 < ROWS_PER_BLOCK / 8) {      // prefetch next key row for this wave
      __builtin_prefetch(keyrow + (size_t)8 * BDIM * DDIM + d0, 0, 0);
    }
    const v4f k0 = ld4_nt(keyrow + d0);
    const v4f k1 = ld4_nt(keyrow + d1);

    float r;
    {
      float px, py, pz, pw;
      // low half (d0..d0+3)
      px = tanhf(k0.x + bs0.x + am1 * m00.x + a0 * m10.x + ap1 * m20.x);
      py = tanhf(k0.y + bs0.y + am1 * m00.y + a0 * m10.y + ap1 * m20.y);
      pz = tanhf(k0.z + bs0.z + am1 * m00.z + a0 * m10.z + ap1 * m20.z);
      pw = tanhf(k0.w + bs0.w + am1 * m00.w + a0 * m10.w + ap1 * m20.w);
      r = px * wa0.x + py * wa0.y + pz * wa0.z + pw * wa0.w;
      // high half (d1..d1+3)
      px = tanhf(k1.x + bs1.x + am1 * m01.x + a0 * m11.x + ap1 * m21.x);
      py = tanhf(k1.y + bs1.y + am1 * m01.y + a0 * m11.y + ap1 * m21.y);
      pz = tanhf(k1.z + bs1.z + am1 * m01.z + a0 * m11.z + ap1 * m21.z);
      pw = tanhf(k1.w + bs1.w + am1 * m01.w + a0 * m11.w + ap1 * m21.w);
      r += px * wa1.x + py * wa1.y + pz * wa1.z + pw * wa1.w;
    }
    // wave32 all-reduce
    #pragma unroll
    for (int off = 16; off >= 1; off >>= 1) r += __shfl_xor(r, off, 32);

    const float score = r + battn;
    const float h = (score >= 0.f) ? mask_s[j] : 0.f;   // unnormalized hard gate
    if (lane == 0) hard_out[(size_t)l * BDIM + b] = h;
    Sacc += h;

    if (h != 0.f) {   // wave-uniform: skip ctx_val traffic for gated-off rows
      const float* valrow = ctx_val + ((size_t)l * BDIM + b) * DDIM;
      const v4f v0 = ld4_nt(valrow + d0);
      const v4f v1 = ld4_nt(valrow + d1);
      acc0.x += h * v0.x; acc0.y += h * v0.y; acc0.z += h * v0.z; acc0.w += h * v0.w;
      acc1.x += h * v1.x; acc1.y += h * v1.y; acc1.z += h * v1.z; acc1.w += h * v1.w;
    }
  }

  // deterministic block reduction across the 8 waves
  *(v4f*)&ctLDS[wave * DDIM + d0] = acc0;
  *(v4f*)&ctLDS[wave * DDIM + d1] = acc1;
  if (lane == 0) SLDS[wave] = Sacc;
  __syncthreads();

  const int d = threadIdx.x;
  float s = 0.f;
  #pragma unroll
  for (int w = 0; w < 8; ++w) s += ctLDS[w * DDIM + d];
  ctp[((size_t)split * BDIM + b) * DDIM + d] = s;
  if (threadIdx.x == 0) {
    float ss = 0.f;
    #pragma unroll
    for (int w = 0; w < 8; ++w) ss += SLDS[w];
    Sp[split * BDIM + b] = ss;
  }
}

// ---------------------------------------------------------------------------
// Kernel 4: normalize. out[0:B*D] = ct, out[B*D : B*D+L*B] = hard (in place)
// ---------------------------------------------------------------------------
__global__ __launch_bounds__(256) void finalize(const float* __restrict__ ctp,
                                                const float* __restrict__ Sp,
                                                float* __restrict__ out) {
  const int idx = blockIdx.x * 256 + threadIdx.x;
  const int NCT = BDIM * DDIM;
  if (idx < NCT) {
    const int b = idx / DDIM;
    float S = 1e-10f;
    #pragma unroll
    for (int s = 0; s < NSPLIT; ++s) S += Sp[s * BDIM + b];
    float acc = 0.f;
    #pragma unroll
    for (int s = 0; s < NSPLIT; ++s) acc += ctp[(size_t)s * NCT + idx];
    out[idx] = acc / S;
  } else if (idx < NCT + LSEQ * BDIM) {
    const int j = idx - NCT;      // l*B + b
    const int b = j & (BDIM - 1);
    float S = 1e-10f;
    #pragma unroll
    for (int s = 0; s < NSPLIT; ++s) S += Sp[s * BDIM + b];
    out[idx] = out[idx] / S;
  }
}

// ---------------------------------------------------------------------------
extern "C" void kernel_launch(void* const* d_in, const int* in_sizes, int n_in,
                              void* d_out, int out_size, void* d_ws, size_t ws_size,
                              hipStream_t stream) {
  const float* ctx_val    = (const float*)d_in[0];
  const float* ctx_key    = (const float*)d_in[1];
  const float* ctx_mask   = (const float*)d_in[2];
  const float* att_past   = (const float*)d_in[3];
  const float* ht_query   = (const float*)d_in[4];
  const float* W_conv     = (const float*)d_in[5];
  const float* W_query    = (const float*)d_in[6];
  const float* W_att_past = (const float*)d_in[7];
  const float* b_att_past = (const float*)d_in[8];
  const float* W_attn     = (const float*)d_in[9];
  const float* b_attn     = (const float*)d_in[10];
  float* out = (float*)d_out;

  float* ws      = (float*)d_ws;
  float* ws_base = ws;                         // B*D           = 16384
  float* ws_M    = ws_base + BDIM * DDIM;      // 3*D           = 768
  float* ws_ctp  = ws_M + 3 * DDIM;            // NSPLIT*B*D    = 131072
  float* ws_Sp   = ws_ctp + NSPLIT * BDIM * DDIM;  // NSPLIT*B  = 512

  query_gemm<<<8, 256, 0, stream>>>(ht_query, W_query, b_att_past, ws_base);
  prep_M<<<1, 256, 0, stream>>>(W_att_past, W_conv, ws_M);

  dim3 g2(NSPLIT, BDIM);
  main_kernel<<<g2, 256, 0, stream>>>(ctx_val, ctx_key, ctx_mask, att_past,
                                      W_attn, b_attn, ws_base, ws_M,
                                      ws_ctp, ws_Sp, out + BDIM * DDIM);

  const int total = BDIM * DDIM + LSEQ * BDIM;
  finalize<<<(total + 255) / 256, 256, 0, stream>>>(ws_ctp, ws_Sp, out);
}